// T_wise_graphattention_19825569038757
// MI455X (gfx1250) — compile-verified
//
#include <hip/hip_runtime.h>
#include <hip/hip_bf16.h>

typedef __attribute__((ext_vector_type(2))) float v2f;
typedef __attribute__((ext_vector_type(8))) float v8f;

#define EHID 64
#define RHID 32
#define DFEAT 160            // 2*EHID + RHID
#define DOUT 224             // EHID + DFEAT
#define EPSF 1e-16f

static inline int ceil_div(int a, int b) { return (a + b - 1) / b; }

// ---------------- kernel 1a: zero output buffer ----------------
__global__ void k_zero_f32(float* p, int n) {
    int i = blockIdx.x * blockDim.x + threadIdx.x;
    if (i < n) p[i] = 0.0f;
}

// ---------------- kernel 1b: zero segmax (uint) + segsum (f32) ----------------
__global__ void k_zero_u32(unsigned* p, int n) {
    int i = blockIdx.x * blockDim.x + threadIdx.x;
    if (i < n) p[i] = 0u;   // 0 is the minimum under the ordered-uint encoding
}

// ---------------- kernel 2: attention scores via WMMA ----------------
// One wave (32 lanes) computes scores for 16 edges.
// A tile (16x4 f32): lane L -> edge m = L&15, khalf = L>>4;
//   VGPR0/1 hold K = 2*khalf, 2*khalf+1 (ISA 7.12.2 layout).
// B tile (4x16 f32): row K broadcast across all 16 columns = w[kbase+K],
//   same (khalf -> K rows {2h,2h+1}) striping as A.
// After 40 chunks, every column of D equals the 16 scores; lane 0 holds
// rows 0..7 (c[0..7]), lane 16 holds rows 8..15.
__global__ void k_att_wmma(const float* __restrict__ x,
                           const int* __restrict__ ei,
                           const int* __restrict__ ej,
                           const int* __restrict__ rel,
                           const float* __restrict__ rel_emb,
                           const float* __restrict__ w,
                           float* __restrict__ att, int E) {
    __shared__ float sw[DFEAT];
    int t = threadIdx.x;
    if (t < DFEAT) sw[t] = w[t];
    __syncthreads();

    int wave  = blockIdx.x * (blockDim.x >> 5) + (t >> 5);
    int lane  = t & 31;
    int ebase = wave << 4;
    if (ebase >= E) return;                 // wave-uniform

    int m     = lane & 15;
    int khalf = lane >> 4;
    int e     = ebase + m;

    int vi = ei[e], vj = ej[e], r = rel[e];
    const float* xi = x + (long)vi * EHID;
    const float* xj = x + (long)vj * EHID;
    const float* re = rel_emb + (long)r * RHID;

    v8f c = {};
#pragma unroll
    for (int ch = 0; ch < DFEAT / 4; ++ch) {
        int k0 = ch * 4 + khalf * 2;        // even; (k0,k0+1) never crosses 64/96
        v2f a;
        if (k0 < EHID)              a = *(const v2f*)(xi + k0);
        else if (k0 < EHID + RHID)  a = *(const v2f*)(re + (k0 - EHID));
        else                        a = *(const v2f*)(xj + (k0 - EHID - RHID));
        v2f b;
        b.x = sw[k0];
        b.y = sw[k0 + 1];
        c = __builtin_amdgcn_wmma_f32_16x16x4_f32(
                /*neg_a=*/false, a, /*neg_b=*/false, b,
                /*c_mod=*/(short)0, c, /*reuse_a=*/false, /*reuse_b=*/false);
    }

    if ((lane & 15) == 0) {                 // lane 0 -> scores 0..7, lane 16 -> 8..15
        float4 lo = make_float4(c[0], c[1], c[2], c[3]);
        float4 hi = make_float4(c[4], c[5], c[6], c[7]);
        float4* o = (float4*)(att + ebase + (khalf << 3));
        o[0] = lo;
        o[1] = hi;
    }
}

// ordered-uint encoding: monotone map f32 -> u32 (handles negatives)
__device__ __forceinline__ unsigned enc_ord(float f) {
    unsigned u = __float_as_uint(f);
    return (u & 0x80000000u) ? ~u : (u | 0x80000000u);
}
__device__ __forceinline__ float dec_ord(unsigned u) {
    return (u & 0x80000000u) ? __uint_as_float(u & 0x7FFFFFFFu)
                             : __uint_as_float(~u);
}

// ---------------- kernel 3: segment max ----------------
__global__ void k_segmax(const float* __restrict__ att, const int* __restrict__ ei,
                         unsigned* __restrict__ segmax_u, int E) {
    int e = blockIdx.x * blockDim.x + threadIdx.x;
    if (e >= E) return;
    atomicMax(&segmax_u[ei[e]], enc_ord(att[e]));
}

// ---------------- kernel 4: exp + segment sum ----------------
__global__ void k_expsum(float* __restrict__ att, const int* __restrict__ ei,
                         const unsigned* __restrict__ segmax_u,
                         float* __restrict__ segsum, int E) {
    int e = blockIdx.x * blockDim.x + threadIdx.x;
    if (e >= E) return;
    int vi = ei[e];
    float p = expf(att[e] - dec_ord(segmax_u[vi]));
    att[e] = p;                             // overwrite score with exp()
    atomicAdd(&segsum[vi], p);
}

// ---------------- kernel 5: weighted scatter-sum ----------------
// One wave per edge; lanes cover d = lane + 32*i (segment choice is
// wave-uniform per i since boundaries 64/96 are multiples of 32).
__global__ void k_scatter(const float* __restrict__ x,
                          const int* __restrict__ ei,
                          const int* __restrict__ ej,
                          const int* __restrict__ rel,
                          const float* __restrict__ rel_emb,
                          const float* __restrict__ p,
                          const float* __restrict__ segsum,
                          float* __restrict__ out, int E) {
    int wave = blockIdx.x * (blockDim.x >> 5) + (threadIdx.x >> 5);
    int lane = threadIdx.x & 31;
    if (wave >= E) return;
    int e = wave;
    int vi = ei[e], vj = ej[e], r = rel[e];
    float coef = p[e] / (segsum[vi] + EPSF);
    float* orow = out + (long)vi * DOUT + EHID;
    const float* xi = x + (long)vi * EHID;
    const float* xj = x + (long)vj * EHID;
    const float* re = rel_emb + (long)r * RHID;
#pragma unroll
    for (int i = 0; i < DFEAT / 32; ++i) {
        int d = lane + 32 * i;
        float v;
        if (d < EHID)              v = xi[d];
        else if (d < EHID + RHID)  v = re[d - EHID];
        else                       v = xj[d - EHID - RHID];
        atomicAdd(orow + d, v * coef);
    }
}

// ---------------- kernel 6: finalize (copy x, ReLU aggregate) ----------------
__global__ void k_finalize(const float* __restrict__ x, float* __restrict__ out, int total) {
    int i = blockIdx.x * blockDim.x + threadIdx.x;
    if (i >= total) return;
    int c = i % DOUT;
    int n = i / DOUT;
    if (c < EHID) out[i] = x[n * EHID + c];
    else          out[i] = fmaxf(out[i], 0.0f);
}

extern "C" void kernel_launch(void* const* d_in, const int* in_sizes, int n_in,
                              void* d_out, int out_size, void* d_ws, size_t ws_size,
                              hipStream_t stream) {
    const float* x        = (const float*)d_in[0];
    const int*   eidx     = (const int*)d_in[1];
    const int*   rel      = (const int*)d_in[2];
    const float* rel_emb  = (const float*)d_in[3];
    const float* w        = (const float*)d_in[4];
    float* out = (float*)d_out;

    const int E = in_sizes[2];              // rel_all has one entry per edge
    const int N = in_sizes[0] / EHID;
    const int* ei = eidx;
    const int* ej = eidx + E;

    // workspace carve-up: att[E] f32 | segmax_u[N] u32 | segsum[N] f32
    char* ws = (char*)d_ws;
    float*    att      = (float*)ws;
    unsigned* segmax_u = (unsigned*)(ws + (size_t)E * 4);
    float*    segsum   = (float*)(ws + (size_t)E * 4 + (size_t)N * 4);

    const int B = 256;

    // 1) zero output + scratch
    int tot_out = N * DOUT;
    k_zero_f32<<<ceil_div(tot_out, B), B, 0, stream>>>(out, tot_out);
    k_zero_u32<<<ceil_div(2 * N, B), B, 0, stream>>>(segmax_u, 2 * N); // segmax + segsum

    // 2) attention scores (WMMA): E/16 waves, 8 waves/block
    int waves = E / 16;
    k_att_wmma<<<ceil_div(waves, 8), B, 0, stream>>>(x, ei, ej, rel, rel_emb, w, att, E);

    // 3) segment max
    k_segmax<<<ceil_div(E, B), B, 0, stream>>>(att, ei, segmax_u, E);

    // 4) exp + segment sum
    k_expsum<<<ceil_div(E, B), B, 0, stream>>>(att, ei, segmax_u, segsum, E);

    // 5) weighted scatter-sum: one wave per edge
    k_scatter<<<ceil_div(E, 8), B, 0, stream>>>(x, ei, ej, rel, rel_emb, att, segsum, out, E);

    // 6) finalize
    k_finalize<<<ceil_div(tot_out, B), B, 0, stream>>>(x, out, tot_out);
}